// MAB_78030965834376
// MI455X (gfx1250) — compile-verified
//
#include <hip/hip_runtime.h>
#include <hip/hip_bf16.h>
#include <math.h>

#define N_NODES 50000
#define S_SEG   50000
#define E_EDGES 1000000
#define DMODEL  128
#define H_HEADS 8
#define DH      16

typedef float v2f __attribute__((ext_vector_type(2)));
typedef float v8f __attribute__((ext_vector_type(8)));

// b128 payload type, exactly as the builtin's prototype prints it
typedef int tdm_v4i __attribute__((vector_size(4 * sizeof(int))));
typedef tdm_v4i __attribute__((address_space(1))) as1_v4i;
typedef tdm_v4i __attribute__((address_space(3))) as3_v4i;

#if defined(__HIP_DEVICE_COMPILE__) &&                                        \
    __has_builtin(__builtin_amdgcn_global_load_async_to_lds_b128) &&          \
    __has_builtin(__builtin_amdgcn_s_wait_asynccnt)
#define USE_ASYNC_LDS 1
#else
#define USE_ASYNC_LDS 0
#endif

// ---------------------------------------------------------------------------
// Cooperatively stage a contiguous 16x128-float tile (8 KB) into LDS.
// 256 threads, 2 x b128 per thread.  Uses the gfx1250 async-DMA path
// (GLOBAL_LOAD_ASYNC_TO_LDS_B128 + s_wait_asynccnt) when the toolchain
// exposes it; otherwise plain load + ds_store.
// ---------------------------------------------------------------------------
__device__ __forceinline__ void stage_tile_8kb(const float* __restrict__ gsrc,
                                               float* __restrict__ lds) {
#if USE_ASYNC_LDS
#pragma unroll
    for (int i = 0; i < 2; ++i) {
        const int q = (int)threadIdx.x + i * 256;          // float4 index
        const unsigned long long ga =
            (unsigned long long)(const void*)(gsrc + q * 4);
        const unsigned la =
            (unsigned)(unsigned long long)(const void*)(lds + q * 4);
        __builtin_amdgcn_global_load_async_to_lds_b128(
            (as1_v4i*)ga, (as3_v4i*)la, 0, 0);
    }
    __builtin_amdgcn_s_wait_asynccnt(0);
    __syncthreads();
#else
#pragma unroll
    for (int i = 0; i < 2; ++i) {
        const int q = (int)threadIdx.x + i * 256;
        ((float4*)lds)[q] = ((const float4*)gsrc)[q];
    }
    __syncthreads();
#endif
}

// ---------------------------------------------------------------------------
// GEMM via V_WMMA_F32_16X16X4_F32:  Out[M,128] = In[M,128] @ W^T + bias
// W is row-major [out_dim=128, in_dim=128] so B[k][n] = W[n*128 + k].
// One wave handles one 16x16 output tile; 8 waves/block cover all 8 N-tiles.
// The shared 16x128 A tile is staged once into LDS (async DMA), then each
// wave reads A fragments via ds_load_b64; B fragments come from L1 (W is hot).
// A frag (16x4 f32): lanes 0-15 hold M=lane, K=kk+{0,1}; lanes 16-31 K=kk+{2,3}.
// B frag (4x16 f32): lanes 0-15 hold N=lane, K=kk+{0,1}; lanes 16-31 K=kk+{2,3}.
// C/D (16x16 f32, 8 VGPRs): vgpr r -> M = r + 8*(lane>=16), N = lane&15.
// ---------------------------------------------------------------------------
__global__ void gemm_bias_kernel(const float* __restrict__ In,
                                 const float* __restrict__ W,
                                 const float* __restrict__ bias,
                                 float* __restrict__ Out) {
    __shared__ float Atile[16 * DMODEL];
    const int mtile = blockIdx.x;
    const int ntile = threadIdx.x >> 5;        // wave id -> N tile
    const int lane  = threadIdx.x & 31;
    const int lrow  = lane & 15;
    const int half  = lane >> 4;
    const int bcol  = ntile * 16 + lrow;       // B col (N)

    stage_tile_8kb(In + (size_t)mtile * 16 * DMODEL, Atile);

    v8f c = {};
#pragma unroll
    for (int kk = 0; kk < DMODEL; kk += 4) {
        const int k0 = kk + 2 * half;
        v2f a = *(const v2f*)(Atile + lrow * DMODEL + k0);
        v2f b = *(const v2f*)(W + bcol * DMODEL + k0);
        c = __builtin_amdgcn_wmma_f32_16x16x4_f32(false, a, false, b,
                                                  (short)0, c, false, false);
    }
    const float bv = bias[bcol];
#pragma unroll
    for (int r = 0; r < 8; ++r) {
        const int orow = mtile * 16 + r + half * 8;
        Out[orow * DMODEL + bcol] = c[r] + bv;
    }
}

// Same GEMM but epilogue: Out = Res + relu(In@W^T + bias)
__global__ void gemm_relu_res_kernel(const float* __restrict__ In,
                                     const float* __restrict__ W,
                                     const float* __restrict__ bias,
                                     const float* __restrict__ Res,
                                     float* __restrict__ Out) {
    __shared__ float Atile[16 * DMODEL];
    const int mtile = blockIdx.x;
    const int ntile = threadIdx.x >> 5;
    const int lane  = threadIdx.x & 31;
    const int lrow  = lane & 15;
    const int half  = lane >> 4;
    const int bcol  = ntile * 16 + lrow;

    stage_tile_8kb(In + (size_t)mtile * 16 * DMODEL, Atile);

    v8f c = {};
#pragma unroll
    for (int kk = 0; kk < DMODEL; kk += 4) {
        const int k0 = kk + 2 * half;
        v2f a = *(const v2f*)(Atile + lrow * DMODEL + k0);
        v2f b = *(const v2f*)(W + bcol * DMODEL + k0);
        c = __builtin_amdgcn_wmma_f32_16x16x4_f32(false, a, false, b,
                                                  (short)0, c, false, false);
    }
    const float bv = bias[bcol];
#pragma unroll
    for (int r = 0; r < 8; ++r) {
        const int orow = mtile * 16 + r + half * 8;
        const float v  = c[r] + bv;
        Out[orow * DMODEL + bcol] = Res[orow * DMODEL + bcol] + fmaxf(v, 0.f);
    }
}

// ---------------------------------------------------------------------------
// init: mx <- -inf, den <- 0, accum <- 0
// ---------------------------------------------------------------------------
__global__ void init_kernel(int* __restrict__ mx_bits,
                            float* __restrict__ den,
                            float* __restrict__ accum) {
    const int i = blockIdx.x * blockDim.x + threadIdx.x;   // covers S*128
    if (i < S_SEG * H_HEADS) {
        mx_bits[i] = 0xFF800000;   // -inf
        den[i]     = 0.f;
    }
    accum[i] = 0.f;
}

// float atomic max via signed-max / unsigned-min trick
__device__ __forceinline__ void atomicMaxFloat(float* addr, float val) {
    const int bits = __float_as_int(val);
    if (bits >= 0) atomicMax((int*)addr, bits);
    else           atomicMin((unsigned int*)addr, (unsigned int)bits);
}

// ---------------------------------------------------------------------------
// Edge pass 1: logits + segment max.  One thread per (edge, head).
// ---------------------------------------------------------------------------
__global__ void edge_logits_kernel(const float* __restrict__ Qp,
                                   const float* __restrict__ Kp,
                                   const int*   __restrict__ eidx,
                                   float* __restrict__ A,
                                   float* __restrict__ mx) {
    const int t = blockIdx.x * blockDim.x + threadIdx.x;   // e*H + h
    const int e = t >> 3;
    const int h = t & 7;
    const int4 ix = ((const int4*)eidx)[e];
    const int qi = ix.x, kj = ix.y, le = ix.w;

    const float4* q4 = (const float4*)(Qp + qi * DMODEL + h * DH);
    const float4* k4 = (const float4*)(Kp + kj * DMODEL + h * DH);
    float s = 0.f;
#pragma unroll
    for (int i = 0; i < 4; ++i) {
        const float4 q = q4[i], k = k4[i];
        s += q.x * k.x + q.y * k.y + q.z * k.z + q.w * k.w;
    }
    s *= 0.08838834764831845f;   // 1/sqrt(DV=128)
    A[t] = s;
    atomicMaxFloat(&mx[le * H_HEADS + h], s);
}

// ---------------------------------------------------------------------------
// Edge pass 2: exp, segment denominator, unnormalized attn*V accumulation.
// ---------------------------------------------------------------------------
__global__ void edge_accum_kernel(const float* __restrict__ A,
                                  const float* __restrict__ mx,
                                  const float* __restrict__ Vp,
                                  const int*   __restrict__ eidx,
                                  float* __restrict__ den,
                                  float* __restrict__ accum) {
    const int t = blockIdx.x * blockDim.x + threadIdx.x;   // e*H + h
    const int e = t >> 3;
    const int h = t & 7;
    const int4 ix = ((const int4*)eidx)[e];
    const int kj = ix.y, le = ix.w;

    const float ae = expf(A[t] - mx[le * H_HEADS + h]);
    atomicAdd(&den[le * H_HEADS + h], ae);

    const float4* v4 = (const float4*)(Vp + kj * DMODEL + h * DH);
    float* acc = accum + le * DMODEL + h * DH;
#pragma unroll
    for (int i = 0; i < 4; ++i) {
        const float4 v = v4[i];
        atomicAdd(acc + 4 * i + 0, ae * v.x);
        atomicAdd(acc + 4 * i + 1, ae * v.y);
        atomicAdd(acc + 4 * i + 2, ae * v.z);
        atomicAdd(acc + 4 * i + 3, ae * v.w);
    }
}

// ---------------------------------------------------------------------------
// Combine (Qp[h0] + QKV/den), (S,H,DH)->(S,DH,H) transpose, LayerNorm0.
// One wave32 per row; lane holds 4 consecutive output columns.
// ---------------------------------------------------------------------------
__global__ void combine_ln0_kernel(const float* __restrict__ Qp,
                                   const float* __restrict__ accum,
                                   const float* __restrict__ den,
                                   const int*   __restrict__ h0,
                                   const float* __restrict__ g0,
                                   const float* __restrict__ b0,
                                   float* __restrict__ O1) {
    const int s    = blockIdx.x * 8 + (threadIdx.x >> 5);
    const int lane = threadIdx.x & 31;
    const int n    = h0[s];

    float vals[4];
#pragma unroll
    for (int i = 0; i < 4; ++i) {
        const int c = lane * 4 + i;         // output col = d*H + h
        const int h = c & 7;
        const int d = c >> 3;
        const int f = h * DH + d;           // source feature index
        const float dn = den[s * H_HEADS + h];
        const float q  = Qp[n * DMODEL + f];
        const float a  = accum[s * DMODEL + f];
        vals[i] = q + (dn > 0.f ? a / dn : 0.f);
    }
    float sum = vals[0] + vals[1] + vals[2] + vals[3];
#pragma unroll
    for (int off = 16; off > 0; off >>= 1) sum += __shfl_xor(sum, off, 32);
    const float mean = sum * (1.f / 128.f);

    float vs = 0.f;
#pragma unroll
    for (int i = 0; i < 4; ++i) { const float dv = vals[i] - mean; vs += dv * dv; }
#pragma unroll
    for (int off = 16; off > 0; off >>= 1) vs += __shfl_xor(vs, off, 32);
    const float inv = 1.0f / sqrtf(vs * (1.f / 128.f) + 1e-5f);

    float4 o;
    const int c0 = lane * 4;
    o.x = (vals[0] - mean) * inv * g0[c0 + 0] + b0[c0 + 0];
    o.y = (vals[1] - mean) * inv * g0[c0 + 1] + b0[c0 + 1];
    o.z = (vals[2] - mean) * inv * g0[c0 + 2] + b0[c0 + 2];
    o.w = (vals[3] - mean) * inv * g0[c0 + 3] + b0[c0 + 3];
    ((float4*)(O1 + s * DMODEL))[lane] = o;
}

// ---------------------------------------------------------------------------
// In-place row LayerNorm (one wave per row).
// ---------------------------------------------------------------------------
__global__ void layernorm_kernel(float* __restrict__ X,
                                 const float* __restrict__ g,
                                 const float* __restrict__ b) {
    const int s    = blockIdx.x * 8 + (threadIdx.x >> 5);
    const int lane = threadIdx.x & 31;
    float4 v = ((const float4*)(X + s * DMODEL))[lane];

    float sum = v.x + v.y + v.z + v.w;
#pragma unroll
    for (int off = 16; off > 0; off >>= 1) sum += __shfl_xor(sum, off, 32);
    const float mean = sum * (1.f / 128.f);

    float vs = (v.x - mean) * (v.x - mean) + (v.y - mean) * (v.y - mean) +
               (v.z - mean) * (v.z - mean) + (v.w - mean) * (v.w - mean);
#pragma unroll
    for (int off = 16; off > 0; off >>= 1) vs += __shfl_xor(vs, off, 32);
    const float inv = 1.0f / sqrtf(vs * (1.f / 128.f) + 1e-5f);

    const int c0 = lane * 4;
    float4 o;
    o.x = (v.x - mean) * inv * g[c0 + 0] + b[c0 + 0];
    o.y = (v.y - mean) * inv * g[c0 + 1] + b[c0 + 1];
    o.z = (v.z - mean) * inv * g[c0 + 2] + b[c0 + 2];
    o.w = (v.w - mean) * inv * g[c0 + 3] + b[c0 + 3];
    ((float4*)(X + s * DMODEL))[lane] = o;
}

// ---------------------------------------------------------------------------
extern "C" void kernel_launch(void* const* d_in, const int* in_sizes, int n_in,
                              void* d_out, int out_size, void* d_ws, size_t ws_size,
                              hipStream_t stream) {
    const float* Q   = (const float*)d_in[0];
    const float* K   = (const float*)d_in[1];
    const int*   h0  = (const int*)  d_in[2];
    const int*   ext = (const int*)  d_in[3];
    const float* Wq  = (const float*)d_in[4];
    const float* bq  = (const float*)d_in[5];
    const float* Wk  = (const float*)d_in[6];
    const float* bk  = (const float*)d_in[7];
    const float* Wv  = (const float*)d_in[8];
    const float* bv  = (const float*)d_in[9];
    const float* Wo  = (const float*)d_in[10];
    const float* bo  = (const float*)d_in[11];
    const float* g0  = (const float*)d_in[12];
    const float* b0  = (const float*)d_in[13];
    const float* g1  = (const float*)d_in[14];
    const float* b1  = (const float*)d_in[15];
    float* out = (float*)d_out;

    // workspace carve-up (floats)
    float* ws    = (float*)d_ws;
    float* Qp    = ws;                                 // N*128
    float* Kp    = Qp    + (size_t)N_NODES * DMODEL;   // N*128
    float* Vp    = Kp    + (size_t)N_NODES * DMODEL;   // N*128
    float* accum = Vp    + (size_t)N_NODES * DMODEL;   // S*128
    float* O1    = accum + (size_t)S_SEG   * DMODEL;   // S*128
    float* A     = O1    + (size_t)S_SEG   * DMODEL;   // E*8
    float* mx    = A     + (size_t)E_EDGES * H_HEADS;  // S*8
    float* den   = mx    + (size_t)S_SEG   * H_HEADS;  // S*8

    const int MT_N = N_NODES / 16;   // 3125
    const int MT_S = S_SEG   / 16;   // 3125

    // 1) projections (WMMA f32)
    gemm_bias_kernel<<<MT_N, 256, 0, stream>>>(Q, Wq, bq, Qp);
    gemm_bias_kernel<<<MT_N, 256, 0, stream>>>(K, Wk, bk, Kp);
    gemm_bias_kernel<<<MT_N, 256, 0, stream>>>(K, Wv, bv, Vp);

    // 2) init segment buffers (S*128 threads, exact)
    init_kernel<<<(S_SEG * DMODEL) / 256, 256, 0, stream>>>((int*)mx, den, accum);

    // 3) edge logits + segment max  (E*H threads, exact)
    edge_logits_kernel<<<(E_EDGES * H_HEADS) / 256, 256, 0, stream>>>(Qp, Kp, ext, A, mx);

    // 4) exp / denominator / weighted-V accumulation
    edge_accum_kernel<<<(E_EDGES * H_HEADS) / 256, 256, 0, stream>>>(A, mx, Vp, ext, den, accum);

    // 5) combine + transpose + LN0  (wave per row)
    combine_ln0_kernel<<<S_SEG / 8, 256, 0, stream>>>(Qp, accum, den, h0, g0, b0, O1);

    // 6) O1 + relu(O1 @ Wo^T + bo)  (WMMA f32) -> d_out
    gemm_relu_res_kernel<<<MT_S, 256, 0, stream>>>(O1, Wo, bo, O1, out);

    // 7) LN1 in place on d_out
    layernorm_kernel<<<S_SEG / 8, 256, 0, stream>>>(out, g1, b1);
}